// yololoss_57836029608363
// MI455X (gfx1250) — compile-verified
//
#include <hip/hip_runtime.h>

#define BS    4096
#define SGRID 14
#define NBOX  2
#define LEN   30          // 5*B + C = 30
#define NCELLS (BS * SGRID * SGRID)

typedef __attribute__((ext_vector_type(2))) float v2f;
typedef __attribute__((ext_vector_type(8))) float v8f;

__global__ void yolo_zero_kernel(float* out) { out[0] = 0.0f; }

__launch_bounds__(256)
__global__ void yolo_loss_kernel(const float* __restrict__ pred,
                                 const float* __restrict__ targ,
                                 float* __restrict__ out)
{
    const float invS = 1.0f / (float)SGRID;

    float acc = 0.0f;

    const int tid    = blockIdx.x * blockDim.x + threadIdx.x;
    const int stride = gridDim.x * blockDim.x;

    for (int cell = tid; cell < NCELLS; cell += stride) {
        // 30 floats per record, 120B => always 8B aligned: stream as v2f, NT.
        const v2f* p2 = reinterpret_cast<const v2f*>(pred + (size_t)cell * LEN);
        const v2f* t2 = reinterpret_cast<const v2f*>(targ + (size_t)cell * LEN);

        float p[LEN], t[LEN];
        #pragma unroll
        for (int k = 0; k < LEN / 2; ++k) {
            v2f a = __builtin_nontemporal_load(&p2[k]);
            v2f b = __builtin_nontemporal_load(&t2[k]);
            p[2 * k] = a[0]; p[2 * k + 1] = a[1];
            t[2 * k] = b[0]; t[2 * k + 1] = b[1];
        }

        const float coo = (t[4] > 0.0f)  ? 1.0f : 0.0f;
        const float noo = (t[4] == 0.0f) ? 1.0f : 0.0f;

        // target box 0 -> xyxy (xy scaled by 1/S, wh not)
        const float tx1 = t[0] * invS - 0.5f * t[2];
        const float ty1 = t[1] * invS - 0.5f * t[3];
        const float tx2 = t[0] * invS + 0.5f * t[2];
        const float ty2 = t[1] * invS + 0.5f * t[3];
        const float area_t = (tx2 - tx1) * (ty2 - ty1);

        float iou[NBOX], pconf[NBOX];
        #pragma unroll
        for (int b = 0; b < NBOX; ++b) {
            const int o = 5 * b;
            const float px1 = p[o] * invS - 0.5f * p[o + 2];
            const float py1 = p[o + 1] * invS - 0.5f * p[o + 3];
            const float px2 = p[o] * invS + 0.5f * p[o + 2];
            const float py2 = p[o + 1] * invS + 0.5f * p[o + 3];
            pconf[b] = p[o + 4];

            const float ltx = fmaxf(px1, tx1);
            const float lty = fmaxf(py1, ty1);
            const float rbx = fminf(px2, tx2);
            const float rby = fminf(py2, ty2);
            const float iw  = fmaxf(rbx - ltx, 0.0f);
            const float ih  = fmaxf(rby - lty, 0.0f);
            const float inter  = iw * ih;
            const float area_p = (px2 - px1) * (py2 - py1);
            iou[b] = inter / (area_p + area_t - inter);
        }

        // argmax with first-max tie-break (box 0 wins ties)
        const float resp0  = (iou[0] >= iou[1]) ? coo : 0.0f;
        const float resp1  = coo - resp0;
        const float maxiou = fmaxf(iou[0], iou[1]);

        float loc = 0.0f, contain = 0.0f, notr = 0.0f, nool = 0.0f;
        #pragma unroll
        for (int b = 0; b < NBOX; ++b) {
            const int o = 5 * b;
            const float respb = (b == 0) ? resp0 : resp1;

            const float dx = p[o]     - t[o];
            const float dy = p[o + 1] - t[o + 1];
            const float sw = sqrtf(p[o + 2]) - sqrtf(t[o + 2]);
            const float sh = sqrtf(p[o + 3]) - sqrtf(t[o + 3]);
            loc += respb * (dx * dx + dy * dy + sw * sw + sh * sh);

            const float dc = pconf[b] - maxiou;
            contain += respb * dc * dc;

            notr += (coo - respb) * pconf[b] * pconf[b];

            const float dn = pconf[b] - t[o + 4];
            nool += noo * dn * dn;
        }

        float cls = 0.0f;
        #pragma unroll
        for (int c = 5 * NBOX; c < LEN; ++c) {
            const float d = p[c] - t[c];
            cls += d * d;
        }
        cls *= coo;

        acc += 5.0f * loc + contain + 0.5f * (notr + nool) + cls;
    }

    // ---- intra-wave reduction via V_WMMA_F32_16X16X4_F32 ----
    // A (16x4 f32, 2 VGPRs): VGPR0 = {A[m][0] lanes 0-15, A[m][2] lanes 16-31},
    // VGPR1 = K=1/K=3 set to 0. B = all-ones 4x16.
    // => D[m][n] = p[m] + p[m+16]. Lane n sums D-VGPRs 0..7 = rows m..m+7;
    // adding the xor-16 partner gives the full 32-lane sum on every lane.
    v2f a;  a[0] = acc;  a[1] = 0.0f;
    v2f bo; bo[0] = 1.0f; bo[1] = 1.0f;
    v8f c = {};
    v8f d = __builtin_amdgcn_wmma_f32_16x16x4_f32(
        false, a, false, bo, (short)0, c, false, false);
    float s = d[0] + d[1] + d[2] + d[3] + d[4] + d[5] + d[6] + d[7];
    s += __shfl_xor(s, 16);

    __shared__ float wsum[8];
    const int lane = threadIdx.x & 31;
    const int wid  = threadIdx.x >> 5;
    if (lane == 0) wsum[wid] = s;
    __syncthreads();

    if (threadIdx.x == 0) {
        float bsum = 0.0f;
        #pragma unroll
        for (int w = 0; w < 8; ++w) bsum += wsum[w];
        atomicAdd(out, bsum * (1.0f / (float)BS));   // loss / N
    }
}

extern "C" void kernel_launch(void* const* d_in, const int* in_sizes, int n_in,
                              void* d_out, int out_size, void* d_ws, size_t ws_size,
                              hipStream_t stream) {
    const float* pred = (const float*)d_in[0];
    const float* targ = (const float*)d_in[1];
    float* out = (float*)d_out;

    yolo_zero_kernel<<<1, 1, 0, stream>>>(out);
    yolo_loss_kernel<<<1024, 256, 0, stream>>>(pred, targ, out);
}